// LSTMModel_6390911336983
// MI455X (gfx1250) — compile-verified
//
#include <hip/hip_runtime.h>

// ---------------------------------------------------------------------------
// 2-layer LSTM (B=4096, T=512, D_IN=1, H=16) + FC(16->1), fp32 throughout.
//
// gates^T[64,16] = W[64,16] @ h^T[16,16] per 16-batch tile via
// V_WMMA_F32_16X16X4_F32 (fp32 precision, K=16 as 4 chained WMMAs).
// One wave32 per tile; 256 waves total -> occupancy is NOT the constraint,
// so we request 1 wave/EU (__launch_bounds__(128,1)) to get the full VGPR
// file per wave and keep ALL weights + (h,c) state in registers (no spills).
// Only global traffic in the loop: one 4B x-read per lane-16 per step
// (entire x = 8MB, L2-resident on 192MB L2).
// D-layout -> B-operand conversion of h uses shfl_xor(16) (no LDS, no bar).
// x rank-1 term + biases fold into the WMMA chains via augmented K=4
// fragments (x at K=0, 1.0 at K=1 multiplying the bias column).
// ---------------------------------------------------------------------------

#define LSTM_B 4096
#define LSTM_T 512

typedef float v2f __attribute__((ext_vector_type(2)));
typedef float v8f __attribute__((ext_vector_type(8)));

static __device__ __forceinline__ v8f wmma4(v2f a, v2f b, v8f c) {
  // D = A(16x4 f32) * B(4x16 f32) + C(16x16 f32)
  return __builtin_amdgcn_wmma_f32_16x16x4_f32(false, a, false, b, (short)0, c,
                                               false, false);
}

static __device__ __forceinline__ float sigm(float x) {
  // 1 / (1 + 2^(-x*log2(e)))  -> v_exp_f32 + v_rcp_f32 (TRANS, co-executes)
  return __builtin_amdgcn_rcpf(1.0f + __builtin_amdgcn_exp2f(-1.44269504f * x));
}
static __device__ __forceinline__ float tanh_a(float x) {
  // tanh(x) = 1 - 2/(1 + e^{2x})
  return 1.0f - 2.0f * __builtin_amdgcn_rcpf(
                           1.0f + __builtin_amdgcn_exp2f(2.88539008f * x));
}

__global__ void __launch_bounds__(128, 1) lstm2_fc_kernel(
    const float* __restrict__ x,     // [B, T]  (D_IN = 1)
    const float* __restrict__ Wih0,  // [64, 1]
    const float* __restrict__ Whh0,  // [64, 16]
    const float* __restrict__ bih0,  // [64]
    const float* __restrict__ bhh0,  // [64]
    const float* __restrict__ Wih1,  // [64, 16]
    const float* __restrict__ Whh1,  // [64, 16]
    const float* __restrict__ bih1,  // [64]
    const float* __restrict__ bhh1,  // [64]
    const float* __restrict__ Wfc,   // [1, 16]
    const float* __restrict__ bfc,   // [1]
    float* __restrict__ out)         // [B, 1]
{
  const int lane = threadIdx.x & 31;
  const int half = lane >> 4;   // which 16-lane half
  const int nIdx = lane & 15;   // N index (batch col of B/C/D, M row of A)
  const int wid  = (blockIdx.x * blockDim.x + threadIdx.x) >> 5;
  const int base = wid * 16;    // batch base of this wave's tile

  // ---- Load weight A-fragments once (A layout: M=lane%16; VGPR0=K{0,2},
  //      VGPR1=K{1,3} split across lane halves -> one float2 load each). ----
  v2f Whh0A[4][4], Wih1A[4][4], Whh1A[4][4];
  v2f Wih0A[4];  // K=0: W_ih0 column, K=1: (b_ih0+b_hh0) column, K=2,3: 0
  v2f B1A[4];    // K=0: (b_ih1+b_hh1) column, K=1..3: 0
#pragma unroll
  for (int j = 0; j < 4; ++j) {
    const int row = 16 * j + nIdx;
#pragma unroll
    for (int kc = 0; kc < 4; ++kc) {
      const float* p0 = Whh0 + row * 16 + kc * 4 + 2 * half;
      const float* p1 = Wih1 + row * 16 + kc * 4 + 2 * half;
      const float* p2 = Whh1 + row * 16 + kc * 4 + 2 * half;
      Whh0A[j][kc] = v2f{p0[0], p0[1]};
      Wih1A[j][kc] = v2f{p1[0], p1[1]};
      Whh1A[j][kc] = v2f{p2[0], p2[1]};
    }
    const float w0  = Wih0[row];
    const float bb0 = bih0[row] + bhh0[row];
    const float bb1 = bih1[row] + bhh1[row];
    Wih0A[j] = v2f{half ? 0.0f : w0, half ? 0.0f : bb0};
    B1A[j]   = v2f{half ? 0.0f : bb1, 0.0f};
  }
  const v2f onesB = v2f{half ? 0.0f : 1.0f, 0.0f};  // K=0 row of ones
  const v8f vzero = {0.f, 0.f, 0.f, 0.f, 0.f, 0.f, 0.f, 0.f};

  // ---- Recurrent state (D layout: M=hidden, N=batch) + B-operand copies ----
  v8f c0 = vzero, h0 = vzero, c1 = vzero, h1 = vzero;
  v2f h0B[4], h1B[4];
#pragma unroll
  for (int kc = 0; kc < 4; ++kc) {
    h0B[kc] = v2f{0.0f, 0.0f};
    h1B[kc] = v2f{0.0f, 0.0f};
  }

  // D-layout tile (M=v+8*halfSrc, N=lane%16) -> B fragments (K rows).
  auto convert = [&](const v8f& h, v2f(&hB)[4]) {
#pragma unroll
    for (int kc = 0; kc < 4; ++kc) {
      const int k0 = 4 * kc;
      if (kc < 2) {  // source rows live in lane half 0
        float sx = __shfl_xor(h[k0 + 2], 16, 32);
        float sy = __shfl_xor(h[k0 + 3], 16, 32);
        hB[kc] = v2f{half ? sx : h[k0 + 0], half ? sy : h[k0 + 1]};
      } else {       // source rows live in lane half 1
        float sx = __shfl_xor(h[k0 - 8], 16, 32);
        float sy = __shfl_xor(h[k0 - 7], 16, 32);
        hB[kc] = v2f{half ? h[k0 - 6] : sx, half ? h[k0 - 5] : sy};
      }
    }
  };

  const float* xrow = x + (base + nIdx) * LSTM_T;

#pragma unroll 1
  for (int t = 0; t < LSTM_T; ++t) {
    // x B-fragment: K=0 row = x[t], K=1 row = 1.0 (bias), K=2,3 = 0
    const float xv = xrow[t];
    const v2f xB = v2f{half ? 0.0f : xv, half ? 0.0f : 1.0f};

    // ---- Layer 0: gates0^T = Wih0*x + bias0 + Whh0 @ h0^T  (5 WMMA/tile)
    v8f g[4];
#pragma unroll
    for (int j = 0; j < 4; ++j) {
      v8f acc = wmma4(Wih0A[j], xB, vzero);
#pragma unroll
      for (int kc = 0; kc < 4; ++kc) acc = wmma4(Whh0A[j][kc], h0B[kc], acc);
      g[j] = acc;
    }
#pragma unroll
    for (int v = 0; v < 8; ++v) {  // gate order: i, f, g, o
      const float iv = sigm(g[0][v]);
      const float fv = sigm(g[1][v]);
      const float gv = tanh_a(g[2][v]);
      const float ov = sigm(g[3][v]);
      const float cv = fv * c0[v] + iv * gv;
      c0[v] = cv;
      h0[v] = ov * tanh_a(cv);
    }
    convert(h0, h0B);

    // ---- Layer 1: gates1^T = bias1 + Wih1 @ h0^T + Whh1 @ h1^T (9 WMMA/tile)
#pragma unroll
    for (int j = 0; j < 4; ++j) {
      v8f acc = wmma4(B1A[j], onesB, vzero);
#pragma unroll
      for (int kc = 0; kc < 4; ++kc) acc = wmma4(Wih1A[j][kc], h0B[kc], acc);
#pragma unroll
      for (int kc = 0; kc < 4; ++kc) acc = wmma4(Whh1A[j][kc], h1B[kc], acc);
      g[j] = acc;
    }
#pragma unroll
    for (int v = 0; v < 8; ++v) {
      const float iv = sigm(g[0][v]);
      const float fv = sigm(g[1][v]);
      const float gv = tanh_a(g[2][v]);
      const float ov = sigm(g[3][v]);
      const float cv = fv * c1[v] + iv * gv;
      c1[v] = cv;
      h1[v] = ov * tanh_a(cv);
    }
    convert(h1, h1B);
  }

  // ---- FC: out[b] = sum_M Wfc[M] * h1(M, b) + bfc ----
  float partial = 0.0f;
#pragma unroll
  for (int v = 0; v < 8; ++v) partial += Wfc[v + 8 * half] * h1[v];
  const float tot = partial + __shfl_xor(partial, 16, 32);
  if (half == 0) out[base + nIdx] = tot + bfc[0];
}

extern "C" void kernel_launch(void* const* d_in, const int* in_sizes, int n_in,
                              void* d_out, int out_size, void* d_ws,
                              size_t ws_size, hipStream_t stream) {
  (void)in_sizes; (void)n_in; (void)d_ws; (void)ws_size; (void)out_size;
  const float* x    = (const float*)d_in[0];
  const float* Wih0 = (const float*)d_in[1];
  const float* Whh0 = (const float*)d_in[2];
  const float* bih0 = (const float*)d_in[3];
  const float* bhh0 = (const float*)d_in[4];
  const float* Wih1 = (const float*)d_in[5];
  const float* Whh1 = (const float*)d_in[6];
  const float* bih1 = (const float*)d_in[7];
  const float* bhh1 = (const float*)d_in[8];
  const float* Wfc  = (const float*)d_in[9];
  const float* bfc  = (const float*)d_in[10];
  float* out = (float*)d_out;

  // 4096 batch rows / 16 per wave = 256 waves; 4 waves (128 thr) per block
  // -> 1 wave per SIMD32 with full VGPR budget (no spills).
  dim3 block(128);
  dim3 grid(LSTM_B / (16 * 4));
  lstm2_fc_kernel<<<grid, block, 0, stream>>>(x, Wih0, Whh0, bih0, bhh0, Wih1,
                                              Whh1, bih1, bhh1, Wfc, bfc, out);
}